// MultiScaleProcessor_38405597561238
// MI455X (gfx1250) — compile-verified
//
#include <hip/hip_runtime.h>
#include <hip/hip_bf16.h>

// Multi-scale bilinear resize + center zero-pad.
// in : [B, 256, 256, 3] f32
// out: [B, 4, 256, 256, 3] f32, scales {32,64,128,256} (ascending), zero-padded.
//
// Memory-bound op (~201 MB stores + ~50 MB loads -> ~11 us @ 23.3 TB/s).
// CDNA5 path used: async global->LDS row staging (ASYNCcnt) + s_wait_asynccnt
// + non-temporal streaming stores for the write-once output.

#define IMG_H 256
#define IMG_W 256
#define NCH   3
#define ROW_FLOATS (IMG_W * NCH)      // 768 floats = 3072 bytes per source row
#define ROW_CHUNKS (ROW_FLOATS / 4)   // 192 x 16B chunks per row

#if defined(__has_builtin)
#  if __has_builtin(__builtin_amdgcn_global_load_async_to_lds_b128)
#    define HAS_ASYNC_LDS 1
#  else
#    define HAS_ASYNC_LDS 0
#  endif
#  if __has_builtin(__builtin_amdgcn_s_wait_asynccnt)
#    define HAS_WAIT_ASYNC 1
#  else
#    define HAS_WAIT_ASYNC 0
#  endif
#else
#  define HAS_ASYNC_LDS 0
#  define HAS_WAIT_ASYNC 0
#endif

typedef int v4i __attribute__((ext_vector_type(4)));
typedef __attribute__((address_space(1))) v4i gv4i;   // global 16B chunk
typedef __attribute__((address_space(3))) v4i lv4i;   // LDS    16B chunk

__global__ __launch_bounds__(256) void
msp_resize_pad_kernel(const float* __restrict__ in, float* __restrict__ out)
{
    __shared__ float rows[2 * ROW_FLOATS];   // 6 KB: row y0 then row y1

    const int x = threadIdx.x;     // output column 0..255
    const int r = blockIdx.x;      // output row    0..255
    const int k = blockIdx.y;      // scale index   0..3
    const int b = blockIdx.z;      // image index

    const int s = 32 << k;         // 32, 64, 128, 256
    const int p = (256 - s) >> 1;  // symmetric pad

    float* orow =
        out + ((((size_t)b * 4 + (size_t)k) * 256 + (size_t)r) * 256 + (size_t)x) * NCH;

    const bool rowIn = (r >= p) && (r < p + s);   // uniform across block
    float wy = 0.0f;

    if (rowIn) {
        const float step = 255.0f / (float)(s - 1);
        const float fy   = fminf((float)(r - p) * step, 255.0f);
        const int   y0   = (int)fy;
        const int   y1   = min(y0 + 1, IMG_H - 1);
        wy = fy - (float)y0;

        float* g0 = const_cast<float*>(in) + ((size_t)b * IMG_H + (size_t)y0) * ROW_FLOATS;
        float* g1 = const_cast<float*>(in) + ((size_t)b * IMG_H + (size_t)y1) * ROW_FLOATS;

#if HAS_ASYNC_LDS
        // 384 16-byte chunks staged via async global->LDS DMA (no VGPR round trip).
        for (int c = x; c < 2 * ROW_CHUNKS; c += 256) {
            float* src = (c < ROW_CHUNKS) ? (g0 + c * 4) : (g1 + (c - ROW_CHUNKS) * 4);
            __builtin_amdgcn_global_load_async_to_lds_b128(
                (gv4i*)src, (lv4i*)&rows[c * 4], /*offset=*/0, /*cpol=*/0);
        }
#if HAS_WAIT_ASYNC
        __builtin_amdgcn_s_wait_asynccnt(0);
#else
        asm volatile("s_wait_asynccnt 0" ::: "memory");
#endif
#else
        // Fallback: vector loads + LDS stores.
        for (int c = x; c < 2 * ROW_CHUNKS; c += 256) {
            const float4 v = (c < ROW_CHUNKS)
                                 ? ((const float4*)g0)[c]
                                 : ((const float4*)g1)[c - ROW_CHUNKS];
            ((float4*)rows)[c] = v;
        }
#endif
    }
    __syncthreads();   // uniform branch above: all threads reach this

    float res[NCH] = {0.0f, 0.0f, 0.0f};

    if (rowIn && x >= p && x < p + s) {
        const float step = 255.0f / (float)(s - 1);
        const float fx   = fminf((float)(x - p) * step, 255.0f);
        const int   x0   = (int)fx;
        const int   x1   = min(x0 + 1, IMG_W - 1);
        const float wx   = fx - (float)x0;

        const float omwx = 1.0f - wx;
        const float omwy = 1.0f - wy;

#pragma unroll
        for (int c = 0; c < NCH; ++c) {
            const float t0 = rows[x0 * NCH + c];
            const float t1 = rows[x1 * NCH + c];
            const float b0 = rows[ROW_FLOATS + x0 * NCH + c];
            const float b1 = rows[ROW_FLOATS + x1 * NCH + c];
            const float top = t0 * omwx + t1 * wx;
            const float bot = b0 * omwx + b1 * wx;
            res[c] = top * omwy + bot * wy;
        }
    }

    // Write-once streaming output: non-temporal stores keep it out of the caches.
    __builtin_nontemporal_store(res[0], orow + 0);
    __builtin_nontemporal_store(res[1], orow + 1);
    __builtin_nontemporal_store(res[2], orow + 2);
}

extern "C" void kernel_launch(void* const* d_in, const int* in_sizes, int n_in,
                              void* d_out, int out_size, void* d_ws, size_t ws_size,
                              hipStream_t stream)
{
    const float* in  = (const float*)d_in[0];
    float*       out = (float*)d_out;

    const int B = in_sizes[0] / (IMG_H * IMG_W * NCH);   // 64 for the reference setup

    dim3 grid(256 /*rows*/, 4 /*scales*/, B /*images*/);
    dim3 block(256, 1, 1);
    msp_resize_pad_kernel<<<grid, block, 0, stream>>>(in, out);
}